// DSBF_51891794870571
// MI455X (gfx1250) — compile-verified
//
#include <hip/hip_runtime.h>

// ---------------------------------------------------------------------------
// STFT -> MVDR-degenerate beamform -> iSTFT for MI455X (gfx1250), fp32 WMMA.
//
// Shapes: x[16,2,320000] f32, steer_re/im[2,257,2] f32.
// N_FFT=512, HOP=128, PAD=256, F=257, T=2501 frames, out_len=320512.
// Outputs (concatenated in d_out):
//   y_time [16,2,2,320000] f32                        (40,960,000 floats)
//   y_beamformed [16,2,257,2501] complex64 interleaved (41,136,448 floats)
//
// Both FFT directions are GEMMs on V_WMMA_F32_16X16X4_F32 (fp32 matrix core,
// full reference precision). Forward basis G fuses Hann * DFT twiddles *
// conj(w) beamform weights (K=1024 spans both channels) -> one GEMM emits
// y_beamformed directly. Inverse basis H fuses irfft scaling * Hann.
// G and H are stored PRE-SWIZZLED in the WMMA B-fragment layout so every
// lane's fragment is one contiguous float2 (global_load_b64, 128B/half-wave).
// Overlap-add is a deterministic gather (no atomics) with precomputed env.
// ---------------------------------------------------------------------------

typedef __attribute__((ext_vector_type(2))) float v2f;
typedef __attribute__((ext_vector_type(8))) float v8f;

#define N_FFT   512
#define HOP     128
#define PADS    256
#define NFREQ   257
#define NFRAMES 2501
#define TSIG    320000
#define OUTLEN  320512          // (2501-1)*128 + 512

// G: K = c*512 + n (1024 rows), col = s*514 + 2f + part (1028, padded 1040).
// Swizzled storage: Gs[row=(K>>2)*2 + khalf][col][j] , K = row*2 + j.
#define G_COLS  1040
#define G_N     (512 * G_COLS * 2)      // 512 swizzled rows * 1040 cols * 2
// H: kk = 2f + part (514 rows, padded to 516), 512 cols, same swizzle.
#define H_COLS  512
#define H_N     (258 * H_COLS * 2)      // 258 swizzled rows * 512 cols * 2
#define ENV_N   OUTLEN
#define Y2_OFF  40960000                // floats: 16*2*2*320000
#define Y2_BS   (NFREQ * NFRAMES * 2)   // per (b,s) stride = 1,285,514
#define FSTRIDE (NFRAMES * 2)           // per-f stride = 5002

#define A_TT    157             // ceil(2501/16) frame tiles (fwd)
#define B_BT    193             // coverage of 2501 frames @ 13 hops/block

// ---------------------------------------------------------------------------
// Init: build G (window*DFT*conj(w)), H (irfft*window), env (Hann^2 OLA sum),
// directly in the swizzled B-fragment layout. Twiddles reduced exactly via
// (f*n) mod 512 then sinpif/cospif.
// ---------------------------------------------------------------------------
__global__ __launch_bounds__(256) void init_ws(const float* __restrict__ sre,
                                               const float* __restrict__ sim,
                                               float* __restrict__ G,
                                               float* __restrict__ H,
                                               float* __restrict__ env) {
  const int total = G_N + H_N + ENV_N;
  for (int idx = blockIdx.x * 256 + threadIdx.x; idx < total;
       idx += gridDim.x * 256) {
    if (idx < G_N) {
      int j = idx & 1, q = idx >> 1;
      int col = q % G_COLS, row = q / G_COLS;      // row = (K>>2)*2 + khalf
      int K = row * 2 + j;                         // K = kg*4 + 2*kh + j
      float v = 0.f;
      if (col < 1028) {
        int c = K >> 9, n = K & 511;
        int s = col / 514, r = col - s * 514, f = r >> 1, part = r & 1;
        int sb = (s * NFREQ + f) * 2;
        float a0r = sre[sb + 0], a0i = sim[sb + 0];
        float a1r = sre[sb + 1], a1i = sim[sb + 1];
        float d = a0r * a0r + a0i * a0i + a1r * a1r + a1i * a1i;
        float wr = (c == 0 ? a0r : a1r) / d;
        float wi = (c == 0 ? a0i : a1i) / d;
        int m = (f * n) & 511;                     // exact mod-512 reduction
        float cth = cospif(m * (1.f / 256.f));
        float sth = sinpif(m * (1.f / 256.f));
        float win = 0.5f - 0.5f * cospif(n * (1.f / 256.f));
        // y_re = sum F*(wr*cos - wi*sin); y_im = sum F*(-wr*sin - wi*cos)
        v = win * (part == 0 ? (wr * cth - wi * sth) : (-wr * sth - wi * cth));
      }
      G[idx] = v;
    } else if (idx < G_N + H_N) {
      int i2 = idx - G_N;
      int j = i2 & 1, q = i2 >> 1;
      int col = q & 511, row = q >> 9;             // 258 swizzled rows
      int K = row * 2 + j;
      float v = 0.f;
      if (K < 514) {
        int f = K >> 1, part = K & 1;
        int m = (f * col) & 511;
        float alpha = (f == 0 || f == 256) ? 1.f : 2.f;
        float win = 0.5f - 0.5f * cospif(col * (1.f / 256.f));
        float sc = alpha * (1.f / 512.f) * win;
        v = (part == 0) ? sc * cospif(m * (1.f / 256.f))
                        : -sc * sinpif(m * (1.f / 256.f));
      }
      H[i2] = v;
    } else {
      int p = idx - G_N - H_N;
      int tlo = (p < 512) ? 0 : ((p - 384) >> 7);
      int thi = p >> 7; if (thi > NFRAMES - 1) thi = NFRAMES - 1;
      float e = 0.f;
      for (int t = tlo; t <= thi; ++t) {
        int n = p - (t << 7);
        float w = 0.5f - 0.5f * cospif(n * (1.f / 256.f));
        e += w * w;
      }
      env[p] = e;
    }
  }
}

// ---------------------------------------------------------------------------
// Kernel A: per (b, 16-frame tile): stage reflect-padded raw samples for both
// channels in LDS (bank-conflict-free padded layout), then GEMM [16 x 1024] x
// G[1024 x 1040] with V_WMMA_F32_16X16X4_F32; D tiles scatter straight into
// the complex y_beamformed output.
// ---------------------------------------------------------------------------
__global__ __launch_bounds__(256) void stft_bf_gemm(const float* __restrict__ x,
                                                    const float* __restrict__ G,
                                                    float* __restrict__ y2) {
  // phys(i) = i + 4*(i>>7): 4-float pad per 128 -> the 16 M-lanes (stride 128)
  // land on distinct LDS banks (stride 132 mod 64 = 4).
  __shared__ float xwin[2][2512];

  int bx = blockIdx.x;
  int b = bx / A_TT, tt = bx - b * A_TT;
  int tid = threadIdx.x;
  int base_p = tt * (16 * HOP);

  for (int i = tid; i < 2 * 2432; i += 256) {
    int c = i / 2432, off = i - c * 2432;
    int j = base_p + off - PADS;                   // reflect padding
    int jr = j < 0 ? -j : (j >= TSIG ? 2 * TSIG - 2 - j : j);
    xwin[c][off + 4 * (off >> 7)] = x[(size_t)(b * 2 + c) * TSIG + jr];
  }
  __syncthreads();

  int wave = tid >> 5, lane = tid & 31;
  int tl = lane & 15, kh = lane >> 4;              // wave32: M-lane, K-half

  for (int nt = wave; nt < 65; nt += 8) {
    v8f acc = {0.f, 0.f, 0.f, 0.f, 0.f, 0.f, 0.f, 0.f};
    int col = nt * 16 + tl;                        // <= 1039, in-bounds
    for (int k0 = 0; k0 < 1024; k0 += 4) {
      int K0 = k0 + 2 * kh;                        // lane's K pair {K0,K0+1}
      int c = k0 >> 9;                             // uniform within group
      int i0 = tl * 128 + (K0 & 511);              // even -> pair stays in-row
      v2f a = *(const v2f*)&xwin[c][i0 + 4 * (i0 >> 7)];
      // swizzled B: one contiguous float2 per lane, 128B per half-wave
      v2f bb = *(const v2f*)&G[((size_t)((k0 >> 1) + kh) * G_COLS + col) * 2];
      acc = __builtin_amdgcn_wmma_f32_16x16x4_f32(false, a, false, bb,
                                                  (short)0, acc, false, false);
    }
    if (col < 1028) {
      int s = col / 514, r = col - s * 514, f = r >> 1, part = r & 1;
      size_t base = (size_t)((b * 2 + s) * NFREQ + f) * FSTRIDE + part;
      int tbase = tt * 16 + 8 * kh;                // D VGPR rr -> M = rr+8*kh
      if (tbase + 8 <= NFRAMES) {                  // fast path: 156/157 tiles
        #pragma unroll
        for (int rr = 0; rr < 8; ++rr)
          y2[base + (size_t)(tbase + rr) * 2] = acc[rr];
      } else {
        #pragma unroll
        for (int rr = 0; rr < 8; ++rr)
          if (tbase + rr < NFRAMES) y2[base + (size_t)(tbase + rr) * 2] = acc[rr];
      }
    }
  }
}

// ---------------------------------------------------------------------------
// Kernel B: per (b,s, 13-hop tile): stage Y[16 frames x 514] in LDS, GEMM with
// H[514 x 512] (irfft*win) via WMMA, park frames back in LDS, then each thread
// gathers the <=4 overlapping frame samples (deterministic OLA), divides by
// env, writes both duplicated output channels coalesced.
// ---------------------------------------------------------------------------
__global__ __launch_bounds__(256) void istft_ola(const float* __restrict__ y2,
                                                 const float* __restrict__ H,
                                                 const float* __restrict__ env,
                                                 float* __restrict__ yt) {
  __shared__ float buf[16 * 524];                  // stride 524: bank-safe

  int bx = blockIdx.x;
  int bs = bx / B_BT, bt = bx - bs * B_BT;
  int tid = threadIdx.x;
  int t0 = bt * 13;
  size_t ybase = (size_t)bs * Y2_BS;

  // Stage A: buf[t][kk] = Y(re/im) for frames t0..t0+15 (zero past end).
  // i -> (kk, t): consecutive threads walk contiguous t (8B stride segments).
  for (int i = tid; i < 516 * 16; i += 256) {
    int kk = i >> 4, t = i & 15;
    int tg = t0 + t;
    float v = 0.f;
    if (kk < 514 && tg < NFRAMES) {
      int f = kk >> 1, part = kk & 1;
      v = y2[ybase + (size_t)f * FSTRIDE + (size_t)tg * 2 + part];
    }
    buf[t * 524 + kk] = v;
  }
  __syncthreads();

  int wave = tid >> 5, lane = tid & 31;
  int tl = lane & 15, kh = lane >> 4;

  v8f acc[4];
  for (int i = 0; i < 4; ++i) {
    int nt = wave * 4 + i;                         // 32 N-tiles of 512 cols
    int col = nt * 16 + tl;
    v8f a8 = {0.f, 0.f, 0.f, 0.f, 0.f, 0.f, 0.f, 0.f};
    for (int k0 = 0; k0 < 516; k0 += 4) {
      int K0 = k0 + 2 * kh;
      v2f a = *(const v2f*)&buf[tl * 524 + K0];
      v2f bb = *(const v2f*)&H[((size_t)((k0 >> 1) + kh) * H_COLS + col) * 2];
      a8 = __builtin_amdgcn_wmma_f32_16x16x4_f32(false, a, false, bb,
                                                 (short)0, a8, false, false);
    }
    acc[i] = a8;
  }
  __syncthreads();

  // Park time-domain frames (already *WIN) back into LDS.
  for (int i = 0; i < 4; ++i) {
    int n = (wave * 4 + i) * 16 + tl;
    #pragma unroll
    for (int rr = 0; rr < 8; ++rr) buf[(rr + 8 * kh) * 524 + n] = acc[i][rr];
  }
  __syncthreads();

  // Deterministic gather overlap-add over owned padded samples.
  int own_lo = (bt == 0) ? 0 : (t0 + 3) * HOP;
  int own_hi = (t0 + 16) * HOP; if (own_hi > OUTLEN) own_hi = OUTLEN;
  for (int p = own_lo + tid; p < own_hi; p += 256) {
    int tlo = (p < 512) ? 0 : ((p - 384) >> 7);
    int ltlo = tlo - t0; if (ltlo < 0) ltlo = 0;
    int lthi = (p >> 7) - t0; if (lthi > 15) lthi = 15;
    float sum = 0.f;
    for (int lt = ltlo; lt <= lthi; ++lt)
      sum += buf[lt * 524 + (p - ((t0 + lt) << 7))];
    if (p >= PADS && p < OUTLEN - PADS) {
      float v = sum / env[p];
      size_t ob = (size_t)(bs * 2) * TSIG + (p - PADS);
      yt[ob] = v;                                  // c = 0
      yt[ob + TSIG] = v;                           // c = 1 (broadcast)
    }
  }
}

// ---------------------------------------------------------------------------
extern "C" void kernel_launch(void* const* d_in, const int* in_sizes, int n_in,
                              void* d_out, int out_size, void* d_ws,
                              size_t ws_size, hipStream_t stream) {
  (void)in_sizes; (void)n_in; (void)out_size; (void)ws_size;
  const float* x   = (const float*)d_in[0];
  const float* sre = (const float*)d_in[1];
  const float* sim = (const float*)d_in[2];
  float* out = (float*)d_out;
  float* ws  = (float*)d_ws;

  float* G   = ws;                       // 1,064,960 floats (swizzled)
  float* H   = G + G_N;                  //   264,192 floats (swizzled)
  float* env = H + H_N;                  //   320,512 floats (~6.6 MB total ws)
  float* y2  = out + Y2_OFF;             // complex y_beamformed region

  const int init_total = G_N + H_N + ENV_N;
  init_ws<<<(init_total + 255) / 256, 256, 0, stream>>>(sre, sim, G, H, env);
  stft_bf_gemm<<<16 * A_TT, 256, 0, stream>>>(x, G, y2);
  istft_ola<<<32 * B_BT, 256, 0, stream>>>(y2, H, env, out);
}